// GNN_1288490189481
// MI455X (gfx1250) — compile-verified
//
#include <hip/hip_runtime.h>
#include <hip/hip_bf16.h>

typedef float v2f __attribute__((ext_vector_type(2)));
typedef float v8f __attribute__((ext_vector_type(8)));

#define DIM 128
#define NEG_SLOPE 0.01f
#define SCAN_T 1024

// ---------------------------------------------------------------------------
// Degree count over destinations (self loop added analytically later).
// ---------------------------------------------------------------------------
__global__ void cnt_kernel(const int* __restrict__ dst, int* __restrict__ cnt, int E) {
    int e = blockIdx.x * blockDim.x + threadIdx.x;
    if (e < E) atomicAdd(&cnt[dst[e]], 1);
}

__global__ void dis_kernel(const int* __restrict__ cnt, float* __restrict__ dis, int N) {
    int i = blockIdx.x * blockDim.x + threadIdx.x;
    if (i < N) dis[i] = rsqrtf((float)cnt[i] + 1.0f);  // +1 = self loop; always > 0
}

// ---------------------------------------------------------------------------
// Exclusive prefix scan of cnt[N] -> row_ptr[N+1], single 1024-thread block
// (Hillis-Steele per chunk with running carry).  N=50K -> ~49 chunks.
// ---------------------------------------------------------------------------
__global__ __launch_bounds__(SCAN_T) void scan_kernel(const int* __restrict__ cnt,
                                                      int* __restrict__ row_ptr, int N) {
    __shared__ int sdata[SCAN_T];
    __shared__ int carry_s;
    if (threadIdx.x == 0) carry_s = 0;
    __syncthreads();
    for (int base = 0; base < N; base += SCAN_T) {
        int i = base + threadIdx.x;
        int v = (i < N) ? cnt[i] : 0;
        sdata[threadIdx.x] = v;
        __syncthreads();
        for (int off = 1; off < SCAN_T; off <<= 1) {
            int t = (threadIdx.x >= off) ? sdata[threadIdx.x - off] : 0;
            __syncthreads();
            sdata[threadIdx.x] += t;
            __syncthreads();
        }
        int incl  = sdata[threadIdx.x];
        int total = sdata[SCAN_T - 1];
        int c     = carry_s;
        if (i < N) row_ptr[i] = c + incl - v;  // exclusive
        __syncthreads();
        if (threadIdx.x == 0) carry_s = c + total;
        __syncthreads();
    }
    if (threadIdx.x == 0) row_ptr[N] = carry_s;
}

__global__ void copy_kernel(const int* __restrict__ srcp, int* __restrict__ dstp, int N) {
    int i = blockIdx.x * blockDim.x + threadIdx.x;
    if (i < N) dstp[i] = srcp[i];
}

// CSR fill: one int atomic per edge; precompute normalized edge weight.
__global__ void fill_csr_kernel(const int* __restrict__ src, const int* __restrict__ dst,
                                const float* __restrict__ dis,
                                int* __restrict__ cursor, int* __restrict__ col,
                                float* __restrict__ val, int E) {
    int e = blockIdx.x * blockDim.x + threadIdx.x;
    if (e >= E) return;
    int s = src[e], d = dst[e];
    int pos = atomicAdd(&cursor[d], 1);
    col[pos] = s;
    val[pos] = dis[s] * dis[d];
}

// ---------------------------------------------------------------------------
// Dense transform  T = A @ W  using fp32 WMMA (V_WMMA_F32_16X16X4_F32).
// Block = 256 threads = 8 wave32s; block stages a 16x128 A stripe in LDS,
// each wave owns one 16x16 output tile.
// ---------------------------------------------------------------------------
__global__ __launch_bounds__(256) void gemm_wmma_kernel(
        const float* __restrict__ A,   // [N, 128]
        const float* __restrict__ W,   // [128, 128]
        float* __restrict__ T,         // [N, 128]
        int N) {
    __shared__ float As[16 * 132];

    const int row0 = blockIdx.x * 16;
    const int tid  = threadIdx.x;

    for (int i = tid; i < 16 * DIM; i += 256) {
        int r = i >> 7;
        int c = i & (DIM - 1);
        int rr = row0 + r;
        As[r * 132 + c] = (rr < N) ? A[(size_t)rr * DIM + c] : 0.0f;
    }
    __syncthreads();

    const int wave  = tid >> 5;
    const int lane  = tid & 31;
    const int m     = lane & 15;
    const int khalf = (lane >> 4) * 2;
    const int n0    = wave * 16;

    v8f acc = {};
#pragma unroll
    for (int k = 0; k < DIM; k += 4) {
        v2f a, b;
        a.x = As[m * 132 + k + khalf];
        a.y = As[m * 132 + k + khalf + 1];
        b.x = W[(k + khalf) * DIM + n0 + m];
        b.y = W[(k + khalf + 1) * DIM + n0 + m];
        acc = __builtin_amdgcn_wmma_f32_16x16x4_f32(
                false, a, false, b, (short)0, acc, false, false);
    }

    const int rbase = row0 + (lane >> 4) * 8;
#pragma unroll
    for (int v = 0; v < 8; ++v) {
        int r = rbase + v;
        if (r < N) T[(size_t)r * DIM + n0 + m] = acc[v];
    }
}

// ---------------------------------------------------------------------------
// Pull-mode SpMM aggregation, fused with self-loop, bias and LeakyReLU.
// One wave32 per node; float4 per lane covers all 128 features.
// Edge metadata (col,val) is batch-loaded 32 edges at a time across lanes,
// then broadcast via __shfl so the per-edge 512B feature gathers are
// independent and can be issued back-to-back (no serial load->gather chain).
// ---------------------------------------------------------------------------
__global__ __launch_bounds__(256) void spmm_kernel(
        const float* __restrict__ T,
        const int*   __restrict__ row_ptr,
        const int*   __restrict__ col,
        const float* __restrict__ val,
        const float* __restrict__ dis,
        const float* __restrict__ bias,
        float*       __restrict__ out,
        int N) {
    int node = blockIdx.x * 8 + (threadIdx.x >> 5);
    if (node >= N) return;
    int lane = threadIdx.x & 31;

    int beg = row_ptr[node];
    int end = row_ptr[node + 1];

    // self loop: weight dis[node]^2
    float ds = dis[node];
    float w0 = ds * ds;
    float4 tv = ((const float4*)(T + (size_t)node * DIM))[lane];
    float4 acc;
    acc.x = tv.x * w0; acc.y = tv.y * w0; acc.z = tv.z * w0; acc.w = tv.w * w0;

    for (int jb = beg; jb < end; jb += 32) {
        int nb = end - jb;              // edges remaining in this batch
        if (nb > 32) nb = 32;
        int   jl  = jb + lane;
        int   s_l = (jl < end) ? col[jl] : 0;
        float w_l = (jl < end) ? val[jl] : 0.0f;
        for (int k = 0; k < nb; ++k) {
            int   s = __shfl(s_l, k);
            float w = __shfl(w_l, k);
            float4 v = ((const float4*)(T + (size_t)s * DIM))[lane];
            acc.x = fmaf(v.x, w, acc.x);
            acc.y = fmaf(v.y, w, acc.y);
            acc.z = fmaf(v.z, w, acc.z);
            acc.w = fmaf(v.w, w, acc.w);
        }
    }

    float4 bv = ((const float4*)bias)[lane];
    float4 r;
    r.x = acc.x + bv.x; r.y = acc.y + bv.y;
    r.z = acc.z + bv.z; r.w = acc.w + bv.w;
    r.x = (r.x >= 0.0f) ? r.x : NEG_SLOPE * r.x;
    r.y = (r.y >= 0.0f) ? r.y : NEG_SLOPE * r.y;
    r.z = (r.z >= 0.0f) ? r.z : NEG_SLOPE * r.z;
    r.w = (r.w >= 0.0f) ? r.w : NEG_SLOPE * r.w;
    ((float4*)(out + (size_t)node * DIM))[lane] = r;
}

// ---------------------------------------------------------------------------
extern "C" void kernel_launch(void* const* d_in, const int* in_sizes, int n_in,
                              void* d_out, int out_size, void* d_ws, size_t ws_size,
                              hipStream_t stream) {
    const float* x  = (const float*)d_in[0];
    const int*   ei = (const int*)  d_in[1];
    const float* W0 = (const float*)d_in[2];
    const float* b0 = (const float*)d_in[3];
    const float* W1 = (const float*)d_in[4];
    const float* b1 = (const float*)d_in[5];
    float* out = (float*)d_out;

    const int N = in_sizes[0] / DIM;
    const int E = in_sizes[1] / 2;
    const int* srcIdx = ei;        // edge_index[0] = message source
    const int* dstIdx = ei + E;    // edge_index[1] = aggregation target

    const size_t NH = (size_t)N * DIM;

    // workspace layout
    float* T       = (float*)d_ws;            // [N,128]
    float* dis     = T + NH;                  // [N]
    float* val     = dis + N;                 // [E]
    int*   cnt     = (int*)(val + E);         // [N]
    int*   row_ptr = cnt + N;                 // [N+1]
    int*   cursor  = row_ptr + (N + 1);       // [N]
    int*   col     = cursor + N;              // [E]

    const int thr   = 256;
    const int gridN = (N + thr - 1) / thr;
    const int gridE = (E + thr - 1) / thr;

    // --- CSR build (shared by both layers; recomputed each call) ---
    hipMemsetAsync(cnt, 0, (size_t)N * sizeof(int), stream);
    cnt_kernel <<<gridE, thr, 0, stream>>>(dstIdx, cnt, E);
    dis_kernel <<<gridN, thr, 0, stream>>>(cnt, dis, N);
    scan_kernel<<<1, SCAN_T, 0, stream>>>(cnt, row_ptr, N);
    copy_kernel<<<gridN, thr, 0, stream>>>(row_ptr, cursor, N);
    fill_csr_kernel<<<gridE, thr, 0, stream>>>(srcIdx, dstIdx, dis, cursor, col, val, E);

    const int gemmGrid = (N + 15) / 16;
    const int nodeGrid = (N + 7) / 8;

    // --- layer 1: h1 = leaky(gcn(x, W0, b0)); h1 lives in d_out ---
    gemm_wmma_kernel<<<gemmGrid, thr, 0, stream>>>(x, W0, T, N);
    spmm_kernel<<<nodeGrid, thr, 0, stream>>>(T, row_ptr, col, val, dis, b0, out, N);

    // --- layer 2: out = leaky(gcn(h1, W1, b1)) ---
    gemm_wmma_kernel<<<gemmGrid, thr, 0, stream>>>(out, W1, T, N);
    spmm_kernel<<<nodeGrid, thr, 0, stream>>>(T, row_ptr, col, val, dis, b1, out, N);
}